// EarthMoversDistance_8804682957211
// MI455X (gfx1250) — compile-verified
//
#include <hip/hip_runtime.h>

// Sinkhorn EMD, separable-kernel formulation, f32 WMMA on gfx1250.
//   K = Ks (x) Ks  with Ks[i][j] = exp(-(i-j)^2)   (EPS = 1)
//   K @ x  (x as 64x64 image)  ==  Ks @ X @ Ks     (Ks symmetric)
// Non-log Sinkhorn (identical arithmetic to reference's log-domain form):
//   EU = (a+TINY) / (Ks@EV@Ks + TINY) ;  EV = (b+TINY) / (Ks@EU@Ks + TINY)
// emd = sum EU o (KsC@EV@Ks + Ks@EV@KsC),  KsC[i][j] = (i-j)^2 * exp(-(i-j)^2)

#if defined(__HIP_DEVICE_COMPILE__)
#if !__has_builtin(__builtin_amdgcn_wmma_f32_16x16x4_f32)
#error "missing __builtin_amdgcn_wmma_f32_16x16x4_f32 on device toolchain"
#endif
#endif

typedef float v2f __attribute__((ext_vector_type(2)));
typedef float v8f __attribute__((ext_vector_type(8)));

#define EMD_TINY 1e-10f
#define MAX_ITER 100

__device__ __forceinline__ float kc_fun(int i, int j) {  // (Ks o cost) entry
    float d = (float)((i - j) * (i - j));
    return d * __expf(-d);
}

__device__ __forceinline__ float blockReduceSum(float v, float* red) {
    const int tid = threadIdx.x;
    __syncthreads();                 // protect `red` from prior use
    red[tid] = v;
    __syncthreads();
    #pragma unroll
    for (int s = 128; s > 0; s >>= 1) {
        if (tid < s) red[tid] += red[tid + s];
        __syncthreads();
    }
    float r = red[0];
    __syncthreads();
    return r;
}

// 64x64x64 GEMM via V_WMMA_F32_16X16X4_F32. 8 waves, 2 output tiles each.
// AMODE/BMODE: 0 = read LDS row-major, 1 = synthesize KsC entry on the fly.
// DOT: instead of storing D, return sum(D o EUw) partial for this thread.
template <int AMODE, int BMODE, bool DOT>
__device__ __forceinline__ float gemm64(const float* __restrict__ A,
                                        const float* __restrict__ B,
                                        float* __restrict__ D,
                                        const float* __restrict__ EUw,
                                        int wave, int lane) {
    const int h  = lane >> 4;   // lane half (WMMA f32 operand layout)
    const int ln = lane & 15;
    float dot = 0.f;
    #pragma unroll
    for (int t = wave; t < 16; t += 8) {
        const int mt = (t >> 2) << 4;
        const int nt = (t & 3) << 4;
        v8f acc = {};
        #pragma unroll
        for (int k0 = 0; k0 < 64; k0 += 4) {
            const int ka = k0 + 2 * h;
            v2f a, b;
            if (AMODE == 0) { a.x = A[(mt + ln) * 64 + ka];
                              a.y = A[(mt + ln) * 64 + ka + 1]; }
            else            { a.x = kc_fun(mt + ln, ka);
                              a.y = kc_fun(mt + ln, ka + 1); }
            if (BMODE == 0) { b.x = B[ka * 64 + nt + ln];
                              b.y = B[(ka + 1) * 64 + nt + ln]; }
            else            { b.x = kc_fun(ka, nt + ln);
                              b.y = kc_fun(ka + 1, nt + ln); }
            acc = __builtin_amdgcn_wmma_f32_16x16x4_f32(
                false, a, false, b, (short)0, acc, false, false);
        }
        #pragma unroll
        for (int r = 0; r < 8; ++r) {
            const int m = mt + r + 8 * h;
            if (DOT) dot += acc[r] * EUw[m * 64 + nt + ln];
            else     D[m * 64 + nt + ln] = acc[r];
        }
    }
    return dot;
}

// One workgroup per (b,c) channel. 256 threads = 8 wave32.
__global__ void __launch_bounds__(256)
sinkhorn_emd_kernel(const float* __restrict__ pred,
                    const float* __restrict__ target,
                    float* __restrict__ chan_out) {
    extern __shared__ float smem[];
    float* Ks = smem;            // 4096: Gibbs kernel exp(-(i-j)^2)
    float* T  = smem + 4096;     // 4096: GEMM temp (also reduction scratch)
    float* EU = smem + 8192;     // 4096: exp(u) image
    float* EV = smem + 12288;    // 4096: exp(v) image

    const int tid  = threadIdx.x;
    const int wave = tid >> 5;
    const int lane = tid & 31;
    const int c    = blockIdx.x;
    const float* p = pred   + (size_t)c * 4096;
    const float* q = target + (size_t)c * 4096;

    // Build Ks; init potentials exp(0)=1.
    for (int i = tid; i < 4096; i += 256) {
        int r = i >> 6, cc = i & 63;
        float d = (float)((r - cc) * (r - cc));
        Ks[i] = __expf(-d);
        EU[i] = 1.f;
        EV[i] = 1.f;
    }

    // Load channel, normalize; keep a+TINY, b+TINY in registers.
    float areg[16], breg[16];
    float sa = 0.f, sb = 0.f;
    #pragma unroll
    for (int k = 0; k < 16; ++k) {
        float av = p[k * 256 + tid], bv = q[k * 256 + tid];
        areg[k] = av; breg[k] = bv;
        sa += av; sb += bv;
    }
    sa = blockReduceSum(sa, T);
    sb = blockReduceSum(sb, T);
    const float ra = 1.f / (sa + EMD_TINY), rb = 1.f / (sb + EMD_TINY);
    #pragma unroll
    for (int k = 0; k < 16; ++k) {
        areg[k] = areg[k] * ra + EMD_TINY;   // exp(log_a)
        breg[k] = breg[k] * rb + EMD_TINY;   // exp(log_b)
    }
    __syncthreads();

    // 100 Sinkhorn iterations, fully resident in LDS.
    for (int it = 0; it < MAX_ITER; ++it) {
        gemm64<0, 0, false>(Ks, EV, T,  nullptr, wave, lane);  __syncthreads();
        gemm64<0, 0, false>(T,  Ks, EU, nullptr, wave, lane);  __syncthreads();
        #pragma unroll
        for (int k = 0; k < 16; ++k) {
            int i = k * 256 + tid;
            EU[i] = areg[k] / (EU[i] + EMD_TINY);
        }
        __syncthreads();
        gemm64<0, 0, false>(Ks, EU, T,  nullptr, wave, lane);  __syncthreads();
        gemm64<0, 0, false>(T,  Ks, EV, nullptr, wave, lane);  __syncthreads();
        #pragma unroll
        for (int k = 0; k < 16; ++k) {
            int i = k * 256 + tid;
            EV[i] = breg[k] / (EV[i] + EMD_TINY);
        }
        __syncthreads();
    }

    // emd_c = sum EU o (KsC@EV@Ks) + sum EU o (Ks@EV@KsC); KsC on the fly.
    float part = 0.f;
    gemm64<1, 0, false>(nullptr, EV, T, nullptr, wave, lane);        __syncthreads();
    part += gemm64<0, 0, true>(T, Ks, nullptr, EU, wave, lane);      __syncthreads();
    gemm64<0, 0, false>(Ks, EV, T, nullptr, wave, lane);             __syncthreads();
    part += gemm64<0, 1, true>(T, nullptr, nullptr, EU, wave, lane); __syncthreads();

    float tot = blockReduceSum(part, T);
    if (tid == 0) chan_out[c] = tot;
}

__global__ void finalize_mean_kernel(const float* __restrict__ chan,
                                     float* __restrict__ out, int nch) {
    float s = 0.f;
    for (int i = threadIdx.x; i < nch; i += 32) s += chan[i];
    #pragma unroll
    for (int off = 16; off > 0; off >>= 1) s += __shfl_down(s, off, 32);
    if (threadIdx.x == 0) out[0] = s / (float)nch;
}

extern "C" void kernel_launch(void* const* d_in, const int* in_sizes, int n_in,
                              void* d_out, int out_size, void* d_ws, size_t ws_size,
                              hipStream_t stream) {
    const float* pred   = (const float*)d_in[0];
    const float* target = (const float*)d_in[1];
    float* chan = (float*)d_ws;
    float* out  = (float*)d_out;

    const int nch = in_sizes[0] / 4096;                 // B*C = 24
    const size_t shmem = (size_t)4 * 4096 * sizeof(float);  // 64 KB

    sinkhorn_emd_kernel<<<nch, 256, shmem, stream>>>(pred, target, chan);
    finalize_mean_kernel<<<1, 32, 0, stream>>>(chan, out, nch);
}